// Net_687194767560
// MI455X (gfx1250) — compile-verified
//
#include <hip/hip_runtime.h>

#define BB 64
#define NN 1024
#define KK 20
#define MM 256
#define NCLS 40
#define HDIM 512
#define EPSV 1e-5f
#define ASTR 66
#define WSTR 66

typedef float v2f __attribute__((ext_vector_type(2)));
typedef float v8f __attribute__((ext_vector_type(8)));

// ---------------- KNN: one thread per query point, cloud staged in LDS ----
__global__ void knn_kernel(const float* __restrict__ pos, int* __restrict__ idx) {
  __shared__ float sx[NN], sy[NN], sz[NN], sq[NN];
  int b = blockIdx.x >> 2;
  int chunk = blockIdx.x & 3;
  const float* pb = pos + (size_t)b * NN * 3;
  for (int j = threadIdx.x; j < NN; j += 256) {
    float x = pb[j*3+0], y = pb[j*3+1], z = pb[j*3+2];
    sx[j]=x; sy[j]=y; sz[j]=z; sq[j] = x*x + y*y + z*z;
  }
  __syncthreads();
  int i = chunk*256 + (int)threadIdx.x;
  float xi = sx[i], yi = sy[i], zi = sz[i], qi = sq[i];
  float bd[KK]; int bi[KK];
  #pragma unroll
  for (int k=0;k<KK;k++){ bd[k]=__builtin_inff(); bi[k]=-1; }
  float wv = __builtin_inff(); int ws = 0;
  for (int j=0;j<NN;j++) {
    if (j==i) continue;
    float d = qi + sq[j] - 2.0f*(xi*sx[j] + yi*sy[j] + zi*sz[j]);
    if (d < wv) {
      bd[ws]=d; bi[ws]=j;
      wv = bd[0]; ws = 0;
      for (int k=1;k<KK;k++) if (bd[k] > wv) { wv=bd[k]; ws=k; }
    }
  }
  int* op = idx + ((size_t)b*NN + i)*KK;
  for (int k=0;k<KK;k++) op[k]=bi[k];
}

// ---------------- FPS: one block per cloud, sequential argmax rounds -------
__global__ void fps_kernel(const float* __restrict__ pos, int* __restrict__ fidx) {
  __shared__ float sx[NN], sy[NN], sz[NN];
  __shared__ float rv[256]; __shared__ int ri[256];
  __shared__ int lastSh;
  int b = blockIdx.x;
  const float* pb = pos + (size_t)b * NN * 3;
  for (int j = threadIdx.x; j < NN; j += 256){ sx[j]=pb[j*3]; sy[j]=pb[j*3+1]; sz[j]=pb[j*3+2]; }
  float dist[4];
  #pragma unroll
  for (int r=0;r<4;r++) dist[r]=1e38f;
  if (threadIdx.x==0){ fidx[b*MM]=0; lastSh=0; }
  __syncthreads();
  for (int it=1; it<MM; it++) {
    int last = lastSh;
    float lx=sx[last], ly=sy[last], lz=sz[last];
    float bv=-1.0f; int bj=NN;
    #pragma unroll
    for (int r=0;r<4;r++){
      int j = (int)threadIdx.x*4 + r;
      float dx=sx[j]-lx, dy=sy[j]-ly, dz=sz[j]-lz;
      float d = dx*dx + dy*dy + dz*dz;
      float nd = fminf(dist[r], d); dist[r]=nd;
      if (nd > bv) { bv=nd; bj=j; }
    }
    rv[threadIdx.x]=bv; ri[threadIdx.x]=bj;
    __syncthreads();
    for (int s=128;s>0;s>>=1){
      if ((int)threadIdx.x < s) {
        float ov = rv[threadIdx.x+s]; int oj = ri[threadIdx.x+s];
        if (ov > rv[threadIdx.x] || (ov==rv[threadIdx.x] && oj<ri[threadIdx.x])) {
          rv[threadIdx.x]=ov; ri[threadIdx.x]=oj;
        }
      }
      __syncthreads();
    }
    if (threadIdx.x==0){ lastSh = ri[0]; fidx[b*MM+it]=ri[0]; }
    __syncthreads();
  }
}

// -------- Pass A: per-feature sum/sumsq of h1=relu(rel.w1+b1), deterministic partials
__global__ void stats1_kernel(const float* __restrict__ pos, const int* __restrict__ idx,
                              const float* __restrict__ w1, const float* __restrict__ b1,
                              float* __restrict__ part1) {
  const int R = BB*NN*KK;
  int f = threadIdx.x & 63;
  int slot = threadIdx.x >> 6;
  float wa = w1[f*3+0], wb = w1[f*3+1], wc = w1[f*3+2], bias = b1[f];
  float acc=0.f, acc2=0.f;
  for (int row = blockIdx.x*4 + slot; row < R; row += gridDim.x*4) {
    int b = row / (NN*KK);
    int rem = row - b*NN*KK;
    int n = rem / KK;
    int j = idx[row];
    const float* pc = pos + ((size_t)b*NN + n)*3;
    const float* pj = pos + ((size_t)b*NN + j)*3;
    float rx = pj[0]-pc[0], ry = pj[1]-pc[1], rz = pj[2]-pc[2];
    float h = fmaf(rx,wa, fmaf(ry,wb, fmaf(rz,wc, bias)));
    h = fmaxf(h, 0.f);
    acc += h; acc2 += h*h;
  }
  __shared__ float rs[256], rq[256];
  rs[threadIdx.x]=acc; rq[threadIdx.x]=acc2;
  __syncthreads();
  if (slot==0) {
    part1[blockIdx.x*128 + f]      = rs[f]+rs[f+64]+rs[f+128]+rs[f+192];
    part1[blockIdx.x*128 + 64 + f] = rq[f]+rq[f+64]+rq[f+128]+rq[f+192];
  }
}

// -------- combine bn1 stats and fold bn1 into w2/b2 ------------------------
__global__ void prep2_kernel(const float* __restrict__ part1, int nblk,
                             const float* __restrict__ g1, const float* __restrict__ be1,
                             const float* __restrict__ w2, const float* __restrict__ b2,
                             float* __restrict__ w2f, float* __restrict__ b2f) {
  __shared__ float s1s[64], t1s[64];
  __shared__ float red[128];
  int t = threadIdx.x; // 128 threads
  float acc=0.f;
  for (int i=0;i<nblk;i++) acc += part1[i*128 + t];
  red[t]=acc;
  __syncthreads();
  if (t<64) {
    float cnt = (float)(BB*NN*KK);
    float m = red[t]/cnt;
    float v = red[t+64]/cnt - m*m;
    float s = g1[t] * rsqrtf(v + EPSV);
    s1s[t] = s; t1s[t] = be1[t] - m*s;
  }
  __syncthreads();
  if (t<64) {
    int g = t;
    float bacc = b2[g];
    for (int f=0;f<64;f++){
      float w = w2[g*64+f];
      bacc += t1s[f]*w;
      w2f[g*64+f] = w * s1s[f];
    }
    b2f[g] = bacc;
  }
}

// -------- Pass B: WMMA f32 16x16x4 GEMM  h2=relu(h1.w2f^T+b2f) + k-max/min + bn2 stats
__global__ void __launch_bounds__(128) edge2_kernel(
    const float* __restrict__ pos, const int* __restrict__ idx,
    const float* __restrict__ w1, const float* __restrict__ b1,
    const float* __restrict__ w2f, const float* __restrict__ b2f,
    float* __restrict__ maxh2, float* __restrict__ minh2,
    float* __restrict__ part2) {
  __shared__ float w2s[64*WSTR];
  __shared__ float As[4][32*ASTR];
  __shared__ float w1s[64*3];
  __shared__ float b1s[64], b2fs[64];
  __shared__ float bsum[4*64], bsq[4*64];
  int tid = threadIdx.x;
  for (int i = tid; i < 64*64; i += 128) { int g=i>>6, f=i&63; w2s[g*WSTR+f] = w2f[i]; }
  for (int i = tid; i < 64*3; i += 128) w1s[i]=w1[i];
  if (tid < 64) { b1s[tid]=b1[tid]; b2fs[tid]=b2f[tid]; }
  for (int i = tid; i < 4*64; i += 128) { bsum[i]=0.f; bsq[i]=0.f; }
  __syncthreads();

  int wave = tid >> 5;
  int lane = tid & 31;
  int l15  = lane & 15;
  int hi   = lane >> 4;           // 0: K pair {0,1} / rows M<8 ; 1: K pair {2,3} / rows M>=8
  float* Aw = As[wave];

  for (int iter=0; iter<16; iter++) {
    int pid = blockIdx.x*64 + iter*4 + wave;   // point id in [0, B*N)
    // ---- build A tile: rows 0..19 = relu(rel.w1+b1); rows 20..31 = 0 ----
    if (lane < KK) {
      int j = idx[(size_t)pid*KK + lane];
      int b = pid >> 10;
      const float* pc = pos + (size_t)pid*3;
      const float* pj = pos + ((size_t)((b<<10) + j))*3;
      float rx=pj[0]-pc[0], ry=pj[1]-pc[1], rz=pj[2]-pc[2];
      for (int f=0; f<64; f++) {
        float h = fmaf(rx, w1s[f*3+0], fmaf(ry, w1s[f*3+1], fmaf(rz, w1s[f*3+2], b1s[f])));
        Aw[lane*ASTR + f] = fmaxf(h, 0.f);
      }
    } else {
      for (int f=0; f<64; f++) Aw[lane*ASTR + f] = 0.f;
    }
    __syncthreads();
    // ---- 2 row-tiles x 4 col-tiles, K=64 via 16 chained V_WMMA_F32_16X16X4_F32 ----
    int fsel = hi*2;
    for (int c=0;c<4;c++) {
      int colg = c*16 + l15;
      float bias = b2fs[colg];
      float cmx = -__builtin_inff(), cmn = __builtin_inff(), csum=0.f, csq=0.f;
      for (int rt=0; rt<2; rt++) {
        v8f acc = {};
        int arow = rt*16 + l15;
        #pragma unroll
        for (int kk=0; kk<16; kk++) {
          v2f a   = *(const v2f*)&Aw [arow*ASTR + kk*4 + fsel];
          v2f bfr = *(const v2f*)&w2s[colg*WSTR + kk*4 + fsel];
          acc = __builtin_amdgcn_wmma_f32_16x16x4_f32(false, a, false, bfr,
                                                      (short)0, acc, false, false);
        }
        int rowbase = rt*16 + hi*8;
        #pragma unroll
        for (int r=0;r<8;r++) {
          int k = rowbase + r;
          if (k < KK) {
            float v = fmaxf(acc[r] + bias, 0.f);
            cmx = fmaxf(cmx, v); cmn = fminf(cmn, v);
            csum += v; csq += v*v;
          }
        }
      }
      float omx = __shfl_xor(cmx, 16, 32);
      float omn = __shfl_xor(cmn, 16, 32);
      float osm = __shfl_xor(csum, 16, 32);
      float osq = __shfl_xor(csq, 16, 32);
      cmx = fmaxf(cmx, omx); cmn = fminf(cmn, omn);
      csum += osm; csq += osq;
      if (hi==0) {
        maxh2[(size_t)pid*64 + colg] = cmx;
        minh2[(size_t)pid*64 + colg] = cmn;
        bsum[wave*64 + colg] += csum;     // unique owner per (wave,col): race-free
        bsq [wave*64 + colg] += csq;
      }
    }
    __syncthreads();
  }
  if (tid < 64) {
    part2[blockIdx.x*128 + tid]      = bsum[tid]+bsum[64+tid]+bsum[128+tid]+bsum[192+tid];
    part2[blockIdx.x*128 + 64 + tid] = bsq [tid]+bsq [64+tid]+bsq [128+tid]+bsq [192+tid];
  }
}

// -------- combine bn2 stats -> scale/shift --------------------------------
__global__ void prep3_kernel(const float* __restrict__ part2, int nblk,
                             const float* __restrict__ g2, const float* __restrict__ be2,
                             float* __restrict__ s2t2) {
  __shared__ float red[128];
  int t = threadIdx.x; // 128
  float acc=0.f;
  for (int i=0;i<nblk;i++) acc += part2[i*128+t];
  red[t]=acc;
  __syncthreads();
  if (t<64) {
    float cnt = (float)(BB*NN*KK);
    float m = red[t]/cnt;
    float v = red[t+64]/cnt - m*m;
    float s = g2[t]*rsqrtf(v+EPSV);
    s2t2[t]    = s;
    s2t2[64+t] = be2[t] - m*s;
  }
}

// -------- gather FPS rows; max over k of bn2(h2) via (max,min,sign) -------
__global__ void gather_kernel(const float* __restrict__ maxh2, const float* __restrict__ minh2,
                              const int* __restrict__ fidx, const float* __restrict__ s2t2,
                              float* __restrict__ fg) {
  int t = blockIdx.x*256 + threadIdx.x;  // B*M*64 = 1,048,576
  int g  = t & 63;
  int bm = t >> 6;
  int b  = bm >> 8;
  int m  = bm & 255;
  int pt = fidx[b*MM + m];
  size_t off = ((size_t)(b*NN + pt))*64 + g;
  float s = s2t2[g], tt = s2t2[64+g];
  fg[t] = (s >= 0.f) ? fmaf(s, maxh2[off], tt) : fmaf(s, minh2[off], tt);
}

// -------- 64->512 linear, max over 256 pts, relu --------------------------
__global__ void __launch_bounds__(512) head_kernel(const float* __restrict__ fg,
    const float* __restrict__ w_nn1, const float* __restrict__ b_nn1,
    float* __restrict__ ymax) {
  __shared__ float ft[128*64];   // 32 KB, two chunks of 128 points
  int b = blockIdx.x;
  int h = threadIdx.x;
  float wr[64];
  #pragma unroll
  for (int f=0; f<64; f++) wr[f] = w_nn1[h*64+f];
  float best = -__builtin_inff();
  for (int chunk=0; chunk<2; chunk++) {
    __syncthreads();
    for (int i = h; i < 128*64; i += 512) ft[i] = fg[(size_t)b*MM*64 + chunk*128*64 + i];
    __syncthreads();
    for (int m=0; m<128; m++) {
      float d = 0.f;
      #pragma unroll
      for (int f=0; f<64; f++) d = fmaf(wr[f], ft[m*64+f], d);
      best = fmaxf(best, d);
    }
  }
  ymax[b*HDIM + h] = fmaxf(best + b_nn1[h], 0.f);
}

// -------- BN over batch axis ----------------------------------------------
__global__ void bn3_kernel(const float* __restrict__ ymax, const float* __restrict__ g,
                           const float* __restrict__ be, float* __restrict__ yn) {
  int h = threadIdx.x;  // 512 threads, 1 block
  float s=0.f, q=0.f;
  for (int b=0;b<BB;b++){ float v = ymax[b*HDIM+h]; s+=v; q+=v*v; }
  float m = s/(float)BB;
  float var = q/(float)BB - m*m;
  float sc = g[h]*rsqrtf(var+EPSV);
  float tt = be[h] - m*sc;
  for (int b=0;b<BB;b++) yn[b*HDIM+h] = fmaf(ymax[b*HDIM+h], sc, tt);
}

// -------- 512->40 head + log_softmax --------------------------------------
__global__ void final_kernel(const float* __restrict__ yn, const float* __restrict__ w4,
                             const float* __restrict__ b4, float* __restrict__ out) {
  __shared__ float yr[HDIM];
  __shared__ float lg[NCLS];
  int b = blockIdx.x;
  int t = threadIdx.x;  // 64
  for (int i=t;i<HDIM;i+=64) yr[i] = yn[b*HDIM+i];
  __syncthreads();
  if (t < NCLS) {
    float d = b4[t];
    for (int f=0; f<HDIM; f++) d = fmaf(yr[f], w4[t*HDIM+f], d);
    lg[t] = d;
  }
  __syncthreads();
  if (t < NCLS) {
    float mx = -__builtin_inff();
    for (int c=0;c<NCLS;c++) mx = fmaxf(mx, lg[c]);
    float se = 0.f;
    for (int c=0;c<NCLS;c++) se += expf(lg[c]-mx);
    out[b*NCLS + t] = (lg[t]-mx) - logf(se);
  }
}

// ---------------------------------------------------------------------------
extern "C" void kernel_launch(void* const* d_in, const int* in_sizes, int n_in,
                              void* d_out, int out_size, void* d_ws, size_t ws_size,
                              hipStream_t stream) {
  (void)in_sizes; (void)n_in; (void)out_size; (void)ws_size;
  const float* pos    = (const float*)d_in[0];
  const float* w1     = (const float*)d_in[1];
  const float* b1     = (const float*)d_in[2];
  const float* g1     = (const float*)d_in[3];
  const float* be1    = (const float*)d_in[4];
  const float* w2     = (const float*)d_in[5];
  const float* b2     = (const float*)d_in[6];
  const float* g2     = (const float*)d_in[7];
  const float* be2    = (const float*)d_in[8];
  const float* w_nn1  = (const float*)d_in[9];
  const float* b_nn1  = (const float*)d_in[10];
  const float* g_bn2  = (const float*)d_in[11];
  const float* be_bn2 = (const float*)d_in[12];
  const float* w_nn4  = (const float*)d_in[13];
  const float* b_nn4  = (const float*)d_in[14];

  char* wsp = (char*)d_ws;
  auto alloc = [&](size_t bytes)->char* {
    char* p = wsp; wsp += (bytes + 255) & ~(size_t)255; return p;
  };
  int*   idx   = (int*)  alloc((size_t)BB*NN*KK*sizeof(int));     // 5.24 MB
  int*   fidx  = (int*)  alloc((size_t)BB*MM*sizeof(int));        // 64 KB
  float* part1 = (float*)alloc((size_t)2048*128*sizeof(float));   // 1 MB
  float* part2 = (float*)alloc((size_t)1024*128*sizeof(float));   // 512 KB
  float* w2f   = (float*)alloc((size_t)64*64*sizeof(float));
  float* b2f   = (float*)alloc((size_t)64*sizeof(float));
  float* s2t2  = (float*)alloc((size_t)128*sizeof(float));
  float* maxh2 = (float*)alloc((size_t)BB*NN*64*sizeof(float));   // 16.8 MB
  float* minh2 = (float*)alloc((size_t)BB*NN*64*sizeof(float));   // 16.8 MB
  float* fg    = (float*)alloc((size_t)BB*MM*64*sizeof(float));   // 4.2 MB
  float* ymax  = (float*)alloc((size_t)BB*HDIM*sizeof(float));
  float* yn    = (float*)alloc((size_t)BB*HDIM*sizeof(float));

  knn_kernel   <<<BB*4, 256, 0, stream>>>(pos, idx);
  fps_kernel   <<<BB,   256, 0, stream>>>(pos, fidx);
  stats1_kernel<<<2048, 256, 0, stream>>>(pos, idx, w1, b1, part1);
  prep2_kernel <<<1,    128, 0, stream>>>(part1, 2048, g1, be1, w2, b2, w2f, b2f);
  edge2_kernel <<<1024, 128, 0, stream>>>(pos, idx, w1, b1, w2f, b2f, maxh2, minh2, part2);
  prep3_kernel <<<1,    128, 0, stream>>>(part2, 1024, g2, be2, s2t2);
  gather_kernel<<<4096, 256, 0, stream>>>(maxh2, minh2, fidx, s2t2, fg);
  head_kernel  <<<BB,   512, 0, stream>>>(fg, w_nn1, b_nn1, ymax);
  bn3_kernel   <<<1,    512, 0, stream>>>(ymax, g_bn2, be_bn2, yn);
  final_kernel <<<BB,    64, 0, stream>>>(yn, w_nn4, b_nn4, (float*)d_out);
}